// GCN_50122268344699
// MI455X (gfx1250) — compile-verified
//
#include <hip/hip_runtime.h>
#include <hip/hip_bf16.h>

// GCN forward on MI455X (gfx1250, wave32, WMMA).
// Pipeline: deg -> dinv -> h=x@W1 (WMMA f32 16x16x4) -> self-loop init + edge
// scatter (f32 atomics, L2-resident) -> h2=relu(agg1+b1)@W2 (WMMA) ->
// self-loop init + scatter into d_out -> +b2, log_softmax (wave32 shuffles).

typedef float v2f __attribute__((ext_vector_type(2)));
typedef float v8f __attribute__((ext_vector_type(8)));

#define N_NODES  50000
#define N_EDGES  1600000
#define D_FEAT   512
#define HIDDEN   128
#define N_CLS    32

// ---------------- degree / normalization ----------------

__global__ void k_init_deg(float* __restrict__ deg) {
    int i = blockIdx.x * blockDim.x + threadIdx.x;
    if (i < N_NODES) deg[i] = 1.0f;   // self-loop contributes 1
}

__global__ void k_count_deg(const int* __restrict__ dst, float* __restrict__ deg) {
    int e = blockIdx.x * blockDim.x + threadIdx.x;
    if (e < N_EDGES) atomicAdd(&deg[dst[e]], 1.0f);
}

__global__ void k_rsqrt(float* __restrict__ deg) {
    int i = blockIdx.x * blockDim.x + threadIdx.x;
    if (i < N_NODES) deg[i] = rsqrtf(deg[i]);
}

// ---------------- GEMM1: h = x @ W1   (M=50000, K=512, N=128) ----------------
// One wave computes one 16x16 f32 tile via V_WMMA_F32_16X16X4_F32.
// Block = 256 threads = 8 waves -> covers N=128. Grid = 50000/16 = 3125.

__global__ void k_gemm1_wmma(const float* __restrict__ x,
                             const float* __restrict__ W1,
                             float* __restrict__ h) {
    const int lane = threadIdx.x & 31;
    const int wave = threadIdx.x >> 5;          // 0..7 -> N tile
    const int row0 = blockIdx.x * 16;           // M tile base
    const int n0   = wave * 16;
    const int mrow = row0 + (lane & 15);        // A: lanes 0-15 M, 16-31 same M
    const int ncol = n0 + (lane & 15);          // B/C/D: N striped across lanes
    const int kb   = (lane >> 4) * 2;           // lanes 0-15: K={0,1}; 16-31: K={2,3}

    const float* xrow = x + (long long)mrow * D_FEAT;
    v8f c = {};
#pragma unroll 8
    for (int k = 0; k < D_FEAT; k += 4) {
        v2f a = *(const v2f*)(xrow + k + kb);                 // aligned float2
        v2f b;
        b.x = W1[(k + kb) * HIDDEN + ncol];
        b.y = W1[(k + kb + 1) * HIDDEN + ncol];
        c = __builtin_amdgcn_wmma_f32_16x16x4_f32(
                false, a, false, b, (short)0, c, false, false);
    }
    // C/D layout: VGPR i -> M = row0 + i + (lane>=16 ? 8 : 0), N = ncol
    const int mbase = row0 + ((lane >> 4) * 8);
#pragma unroll
    for (int i = 0; i < 8; ++i)
        h[(long long)(mbase + i) * HIDDEN + ncol] = c[i];
}

// ---------------- self-loop init: agg[m][:] = h[m][:] * dinv[m]^2 ------------

__global__ void k_selfloop(const float* __restrict__ h,
                           const float* __restrict__ dinv,
                           float* __restrict__ agg, int cols4 /* cols/4 */) {
    int idx = blockIdx.x * blockDim.x + threadIdx.x;     // one float4 each
    int total = N_NODES * cols4;
    if (idx >= total) return;
    int m = idx / cols4;
    int c = (idx - m * cols4) * 4;
    float s = dinv[m]; s *= s;
    const float4 v = *(const float4*)(h + (long long)m * (cols4 * 4) + c);
    float4 o = make_float4(v.x * s, v.y * s, v.z * s, v.w * s);
    *(float4*)(agg + (long long)m * (cols4 * 4) + c) = o;
}

// ---------------- edge scatter: agg[dst] += h[src] * norm --------------------
// LPE lanes per edge, each lane handles 4 consecutive floats (float4 gather,
// 4x global_atomic_add_f32). h/agg are L2-resident (25.6 MB / 6.4 MB).

template <int COLS, int LPE>   // LPE = COLS/4
__global__ void k_scatter(const int* __restrict__ src,
                          const int* __restrict__ dst,
                          const float* __restrict__ dinv,
                          const float* __restrict__ h,
                          float* __restrict__ agg) {
    int idx = blockIdx.x * blockDim.x + threadIdx.x;
    if (idx >= N_EDGES * LPE) return;
    int e  = idx / LPE;
    int c  = (idx - e * LPE) * 4;
    int s  = src[e], d = dst[e];
    float nm = dinv[s] * dinv[d];
    const float4 v = *(const float4*)(h + (long long)s * COLS + c);
    float* out = agg + (long long)d * COLS + c;
    atomicAdd(out + 0, v.x * nm);
    atomicAdd(out + 1, v.y * nm);
    atomicAdd(out + 2, v.z * nm);
    atomicAdd(out + 3, v.w * nm);
}

// ---------------- GEMM2: h2 = relu(agg1 + b1) @ W2  (K=128, N=32) ------------
// Block = 64 threads = 2 waves -> N=32. ReLU + bias fused into A-fragment.

__global__ void k_gemm2_wmma(const float* __restrict__ agg1,
                             const float* __restrict__ b1,
                             const float* __restrict__ W2,
                             float* __restrict__ h2) {
    const int lane = threadIdx.x & 31;
    const int wave = threadIdx.x >> 5;          // 0..1
    const int row0 = blockIdx.x * 16;
    const int n0   = wave * 16;
    const int mrow = row0 + (lane & 15);
    const int ncol = n0 + (lane & 15);
    const int kb   = (lane >> 4) * 2;

    const float* arow = agg1 + (long long)mrow * HIDDEN;
    v8f c = {};
#pragma unroll 8
    for (int k = 0; k < HIDDEN; k += 4) {
        v2f a;
        a.x = fmaxf(arow[k + kb]     + b1[k + kb],     0.0f);
        a.y = fmaxf(arow[k + kb + 1] + b1[k + kb + 1], 0.0f);
        v2f b;
        b.x = W2[(k + kb) * N_CLS + ncol];
        b.y = W2[(k + kb + 1) * N_CLS + ncol];
        c = __builtin_amdgcn_wmma_f32_16x16x4_f32(
                false, a, false, b, (short)0, c, false, false);
    }
    const int mbase = row0 + ((lane >> 4) * 8);
#pragma unroll
    for (int i = 0; i < 8; ++i)
        h2[(long long)(mbase + i) * N_CLS + ncol] = c[i];
}

// ---------------- +b2 and log_softmax (wave32: one lane per class) -----------

__global__ void k_logsoftmax(float* __restrict__ out, const float* __restrict__ b2) {
    int lane = threadIdx.x & 31;
    int row  = blockIdx.x * (blockDim.x >> 5) + (threadIdx.x >> 5);
    if (row >= N_NODES) return;
    float v = out[(long long)row * N_CLS + lane] + b2[lane];
    float m = v;
#pragma unroll
    for (int off = 16; off > 0; off >>= 1)
        m = fmaxf(m, __shfl_xor(m, off, 32));
    float e = __expf(v - m);
    float s = e;
#pragma unroll
    for (int off = 16; off > 0; off >>= 1)
        s += __shfl_xor(s, off, 32);
    out[(long long)row * N_CLS + lane] = v - m - __logf(s);
}

// ---------------- launch ----------------

extern "C" void kernel_launch(void* const* d_in, const int* in_sizes, int n_in,
                              void* d_out, int out_size, void* d_ws, size_t ws_size,
                              hipStream_t stream) {
    const float* x   = (const float*)d_in[0];
    const int*   ei  = (const int*)d_in[1];        // [2, E] (int32)
    const float* W1  = (const float*)d_in[2];
    const float* b1  = (const float*)d_in[3];
    const float* W2  = (const float*)d_in[4];
    const float* b2  = (const float*)d_in[5];
    const int* src = ei;
    const int* dst = ei + N_EDGES;
    float* logits = (float*)d_out;                 // [N, 32]

    // workspace layout (floats), 16B-aligned chunks
    float* dinv = (float*)d_ws;                                // 50000  (deg -> dinv in place)
    float* h    = dinv + 50176;                                // 50000*128
    float* agg1 = h    + (long long)N_NODES * HIDDEN;          // 50000*128
    float* h2   = agg1 + (long long)N_NODES * HIDDEN;          // 50000*32
    (void)in_sizes; (void)n_in; (void)out_size; (void)ws_size;

    // 1) degrees (with self-loop) and dinv = rsqrt(deg)
    k_init_deg <<<(N_NODES + 255) / 256, 256, 0, stream>>>(dinv);
    k_count_deg<<<(N_EDGES + 255) / 256, 256, 0, stream>>>(dst, dinv);
    k_rsqrt    <<<(N_NODES + 255) / 256, 256, 0, stream>>>(dinv);

    // 2) h = x @ W1
    k_gemm1_wmma<<<N_NODES / 16, 256, 0, stream>>>(x, W1, h);

    // 3) agg1 = self-loop term, then scatter-add edges
    k_selfloop<<<(N_NODES * (HIDDEN / 4) + 255) / 256, 256, 0, stream>>>(h, dinv, agg1, HIDDEN / 4);
    k_scatter<HIDDEN, HIDDEN / 4><<<(N_EDGES * (HIDDEN / 4) + 255) / 256, 256, 0, stream>>>(
        src, dst, dinv, h, agg1);

    // 4) h2 = relu(agg1 + b1) @ W2
    k_gemm2_wmma<<<N_NODES / 16, 64, 0, stream>>>(agg1, b1, W2, h2);

    // 5) logits = self-loop term, then scatter-add edges
    k_selfloop<<<(N_NODES * (N_CLS / 4) + 255) / 256, 256, 0, stream>>>(h2, dinv, logits, N_CLS / 4);
    k_scatter<N_CLS, N_CLS / 4><<<(N_EDGES * (N_CLS / 4) + 255) / 256, 256, 0, stream>>>(
        src, dst, dinv, h2, logits);

    // 6) + b2, log_softmax (8 rows per 256-thread block; 50000/8 = 6250 exact)
    k_logsoftmax<<<N_NODES / 8, 256, 0, stream>>>(logits, b2);
}